// GNNModel_67989332296056
// MI455X (gfx1250) — compile-verified
//
#include <hip/hip_runtime.h>
#include <hip/hip_bf16.h>

typedef __attribute__((ext_vector_type(2))) float v2f;
typedef __attribute__((ext_vector_type(8))) float v8f;

#define F_IN 128
#define NGRAPH 1024

// ---------------------------------------------------------------------------
// Degree / normalization
// ---------------------------------------------------------------------------
__global__ void degree_kernel(const int* __restrict__ dst, float* __restrict__ deg, int E) {
    int e = blockIdx.x * blockDim.x + threadIdx.x;
    if (e < E) atomicAdd(&deg[dst[e]], 1.0f);
}

__global__ void dinv_kernel(float* __restrict__ d, int N) {
    int i = blockIdx.x * blockDim.x + threadIdx.x;
    if (i < N) d[i] = __frsqrt_rn(d[i] + 1.0f);   // rsqrt(deg + self-loop)
}

// ---------------------------------------------------------------------------
// WMMA f32 GEMM:  Hout[N x FO] = A[N x K] @ W[K x FO]
// blockDim = (32,4). One block = 16 rows of A staged in LDS; each wave owns
// 16x16 output tiles (looping over column tiles when FO/16 > 4).
// Uses V_WMMA_F32_16X16X4_F32 (full f32 precision, matches reference).
// A fragment (16x4 f32): lanes 0-15 -> M=lane, VGPR0=K0,VGPR1=K1;
//                        lanes 16-31 -> M=lane-16, VGPR0=K2,VGPR1=K3.
// B fragment (4x16 f32): lanes 0-15 -> N=lane, rows K0/K1; lanes16-31 rows K2/K3.
// C/D (16x16 f32): VGPR v -> (M=v, N=lane) for lanes 0-15, (M=v+8) for 16-31.
// ---------------------------------------------------------------------------
template <int K, int FO>
__global__ __launch_bounds__(128) void gemm_wmma_f32(const float* __restrict__ A,
                                                     const float* __restrict__ W,
                                                     float* __restrict__ Hout,
                                                     int nrows) {
    __shared__ float As[16 * K];
    const int row0 = blockIdx.x * 16;
    if (row0 + 16 > nrows) return;               // whole-block guard (uniform)

    const int tid = threadIdx.y * 32 + threadIdx.x;
    for (int i = tid; i < 16 * K; i += 128)      // 16 contiguous rows of A
        As[i] = A[(size_t)row0 * K + i];
    __syncthreads();

    const int lane = threadIdx.x;
    const int half = lane >> 4;                  // 0: lanes 0-15, 1: lanes 16-31
    const int l15  = lane & 15;
    constexpr int NCT = FO / 16;

    for (int ct = threadIdx.y; ct < NCT; ct += 4) {
        const int col0 = ct * 16;
        v8f acc = {};
        for (int k0 = 0; k0 < K; k0 += 4) {
            const int ka = k0 + 2 * half;
            v2f a, b;
            a.x = As[l15 * K + ka + 0];
            a.y = As[l15 * K + ka + 1];
            b.x = W[(size_t)(ka + 0) * FO + col0 + l15];
            b.y = W[(size_t)(ka + 1) * FO + col0 + l15];
            acc = __builtin_amdgcn_wmma_f32_16x16x4_f32(
                false, a, false, b, (short)0, acc, false, false);
        }
        const int mbase = half * 8;
#pragma unroll
        for (int v = 0; v < 8; ++v)
            Hout[(size_t)(row0 + mbase + v) * FO + col0 + l15] = acc[v];
    }
}

// Scalar tail for rows not covered by full 16-row tiles (N%16; unused for N=100000).
template <int K, int FO>
__global__ void gemm_tail(const float* __restrict__ A, const float* __restrict__ W,
                          float* __restrict__ Hout, int rowStart, int nrows) {
    int idx = blockIdx.x * blockDim.x + threadIdx.x;
    int r = rowStart + idx / FO, c = idx % FO;
    if (r >= nrows) return;
    float acc = 0.f;
    for (int k = 0; k < K; ++k) acc += A[(size_t)r * K + k] * W[(size_t)k * FO + c];
    Hout[(size_t)r * FO + c] = acc;
}

// ---------------------------------------------------------------------------
// Edge scatter: agg[dst] += h[src] * dinv[src]*dinv[dst]
// one thread per (edge, 4-channel group); float4 gather + 4 f32 atomics
// ---------------------------------------------------------------------------
template <int FO>
__global__ void scatter_edges(const float* __restrict__ h, const int* __restrict__ src,
                              const int* __restrict__ dst, const float* __restrict__ dinv,
                              float* __restrict__ agg, int E) {
    constexpr int QF = FO / 4;
    long long tid = (long long)blockIdx.x * blockDim.x + threadIdx.x;
    if (tid >= (long long)E * QF) return;
    int e = (int)(tid / QF), fq = (int)(tid % QF);
    int s = src[e], d = dst[e];
    float c = dinv[s] * dinv[d];
    float4 v = *(const float4*)(h + (size_t)s * FO + fq * 4);
    float* ap = agg + (size_t)d * FO + fq * 4;
    atomicAdd(ap + 0, v.x * c);
    atomicAdd(ap + 1, v.y * c);
    atomicAdd(ap + 2, v.z * c);
    atomicAdd(ap + 3, v.w * c);
}

// out (in place over agg) = [relu](agg + h*dinv^2 + bias)
template <int FO, bool RELU>
__global__ void post_kernel(float* __restrict__ agg, const float* __restrict__ h,
                            const float* __restrict__ dinv, const float* __restrict__ bias,
                            int N) {
    long long idx = (long long)blockIdx.x * blockDim.x + threadIdx.x;
    if (idx >= (long long)N * FO) return;
    int n = (int)(idx / FO), f = (int)(idx % FO);
    float di = dinv[n];
    float v = agg[idx] + h[idx] * di * di + bias[f];
    if (RELU) v = fmaxf(v, 0.0f);
    agg[idx] = v;
}

// ---------------------------------------------------------------------------
// Global mean pool + final linear
// ---------------------------------------------------------------------------
__global__ void count_kernel(const int* __restrict__ batch, float* __restrict__ cnt, int N) {
    int i = blockIdx.x * blockDim.x + threadIdx.x;
    if (i < N) atomicAdd(&cnt[batch[i]], 1.0f);
}

__global__ void poolsum_kernel(const float* __restrict__ h, const int* __restrict__ batch,
                               float* __restrict__ pooled, int N) {
    long long tid = (long long)blockIdx.x * blockDim.x + threadIdx.x;
    if (tid >= (long long)N * 32) return;        // 128 channels / 4
    int n = (int)(tid / 32), fq = (int)(tid % 32);
    int g = batch[n];
    float4 v = *(const float4*)(h + (size_t)n * 128 + fq * 4);
    float* pp = pooled + (size_t)g * 128 + fq * 4;
    atomicAdd(pp + 0, v.x);
    atomicAdd(pp + 1, v.y);
    atomicAdd(pp + 2, v.z);
    atomicAdd(pp + 3, v.w);
}

__global__ void final_kernel(const float* __restrict__ pooled, const float* __restrict__ cnt,
                             const float* __restrict__ Wl, const float* __restrict__ bl,
                             float* __restrict__ out, int G) {
    int g = blockIdx.x * blockDim.x + threadIdx.x;
    if (g >= G) return;
    float acc = 0.f;
#pragma unroll 4
    for (int f = 0; f < 128; ++f) acc += pooled[(size_t)g * 128 + f] * Wl[f];
    out[g] = acc / fmaxf(cnt[g], 1.0f) + bl[0];
}

// ---------------------------------------------------------------------------
extern "C" void kernel_launch(void* const* d_in, const int* in_sizes, int n_in,
                              void* d_out, int out_size, void* d_ws, size_t ws_size,
                              hipStream_t stream) {
    const float* x     = (const float*)d_in[0];
    const int*   eidx  = (const int*)d_in[1];
    const int*   batch = (const int*)d_in[2];
    const float* W1 = (const float*)d_in[3];
    const float* b1 = (const float*)d_in[4];
    const float* W2 = (const float*)d_in[5];
    const float* b2 = (const float*)d_in[6];
    const float* W3 = (const float*)d_in[7];
    const float* b3 = (const float*)d_in[8];
    const float* Wl = (const float*)d_in[9];
    const float* bl = (const float*)d_in[10];
    float* out = (float*)d_out;

    const int N = in_sizes[0] / F_IN;     // 100000
    const int E = in_sizes[1] / 2;        // 1600000
    const int* src = eidx;
    const int* dst = eidx + E;

    // workspace carve-out
    char* ws = (char*)d_ws;
    size_t off = 0;
    auto carve = [&](size_t bytes) -> float* {
        float* p = (float*)(ws + off);
        off += (bytes + 255) & ~(size_t)255;
        return p;
    };
    float* dinv   = carve((size_t)N * 4);            // deg -> dinv (in place)
    float* hbuf   = carve((size_t)N * 128 * 4);      // pre-scatter features
    float* abuf   = carve((size_t)N * 128 * 4);      // agg / layer output
    float* pooled = carve((size_t)NGRAPH * 128 * 4);
    float* cnt    = carve((size_t)NGRAPH * 4);

    const int T = 256;
    const int ntiles = N / 16;
    const int nrem = N - ntiles * 16;
    dim3 gblk(32, 4);

    // ---- degree + symmetric normalization ----
    hipMemsetAsync(dinv, 0, (size_t)N * 4, stream);
    degree_kernel<<<(E + T - 1) / T, T, 0, stream>>>(dst, dinv, E);
    dinv_kernel<<<(N + T - 1) / T, T, 0, stream>>>(dinv, N);

    // ---- Layer 1: x(Nx128) @ W1 -> 32ch, scatter, relu ----
    gemm_wmma_f32<128, 32><<<ntiles, gblk, 0, stream>>>(x, W1, hbuf, N);
    if (nrem) gemm_tail<128, 32><<<(nrem * 32 + T - 1) / T, T, 0, stream>>>(x, W1, hbuf, ntiles * 16, N);
    hipMemsetAsync(abuf, 0, (size_t)N * 32 * 4, stream);
    scatter_edges<32><<<(int)(((long long)E * 8 + T - 1) / T), T, 0, stream>>>(hbuf, src, dst, dinv, abuf, E);
    post_kernel<32, true><<<(int)(((long long)N * 32 + T - 1) / T), T, 0, stream>>>(abuf, hbuf, dinv, b1, N);

    // ---- Layer 2: abuf(Nx32) @ W2 -> 64ch, scatter, relu ----
    gemm_wmma_f32<32, 64><<<ntiles, gblk, 0, stream>>>(abuf, W2, hbuf, N);
    if (nrem) gemm_tail<32, 64><<<(nrem * 64 + T - 1) / T, T, 0, stream>>>(abuf, W2, hbuf, ntiles * 16, N);
    hipMemsetAsync(abuf, 0, (size_t)N * 64 * 4, stream);
    scatter_edges<64><<<(int)(((long long)E * 16 + T - 1) / T), T, 0, stream>>>(hbuf, src, dst, dinv, abuf, E);
    post_kernel<64, true><<<(int)(((long long)N * 64 + T - 1) / T), T, 0, stream>>>(abuf, hbuf, dinv, b2, N);

    // ---- Layer 3: abuf(Nx64) @ W3 -> 128ch, scatter, no relu ----
    gemm_wmma_f32<64, 128><<<ntiles, gblk, 0, stream>>>(abuf, W3, hbuf, N);
    if (nrem) gemm_tail<64, 128><<<(nrem * 128 + T - 1) / T, T, 0, stream>>>(abuf, W3, hbuf, ntiles * 16, N);
    hipMemsetAsync(abuf, 0, (size_t)N * 128 * 4, stream);
    scatter_edges<128><<<(int)(((long long)E * 32 + T - 1) / T), T, 0, stream>>>(hbuf, src, dst, dinv, abuf, E);
    post_kernel<128, false><<<(int)(((long long)N * 128 + T - 1) / T), T, 0, stream>>>(abuf, hbuf, dinv, b3, N);

    // ---- mean pool + final linear ----
    hipMemsetAsync(pooled, 0, (size_t)NGRAPH * 128 * 4, stream);
    hipMemsetAsync(cnt, 0, (size_t)NGRAPH * 4, stream);
    count_kernel<<<(N + T - 1) / T, T, 0, stream>>>(batch, cnt, N);
    poolsum_kernel<<<(int)(((long long)N * 32 + T - 1) / T), T, 0, stream>>>(abuf, batch, pooled, N);
    final_kernel<<<(NGRAPH + T - 1) / T, T, 0, stream>>>(pooled, cnt, Wl, bl, out, NGRAPH);
}